// LNN_77945066488232
// MI455X (gfx1250) — compile-verified
//
#include <hip/hip_runtime.h>
#include <hip/hip_bf16.h>

// ---------------------------------------------------------------------------
// Lagrangian-network dynamics via forward-mode jet propagation.
// Per batch row: jet = (value, 8 grads, 36 sym. 2nd derivs) per hidden unit.
// Linear layers  -> fp32 WMMA GEMM  (M=48 padded channels, K=N=512)
// softplus / LN  -> closed-form jet rules (pointwise + wave reductions)
// Final          -> assemble 8x8 Hessian, solve 4x4 system.
// ---------------------------------------------------------------------------

#define HDIM   512
#define LDSK   516     // padded LDS row stride (floats): row delta = 4 banks
#define NCH    45      // 1 + 8 + 36 jet channels
#define MPAD   48      // channels padded to 3 WMMA M-tiles
#define LNEPS  1e-5f

typedef float v2f __attribute__((ext_vector_type(2)));
typedef float v8f __attribute__((ext_vector_type(8)));

#if defined(__HIP_DEVICE_COMPILE__)
#if !__has_builtin(__builtin_amdgcn_wmma_f32_16x16x4_f32)
#error "gfx1250 f32 WMMA builtin not available"
#endif
#endif

// pair tables only used where the channel index is a *runtime* value
__constant__ int g_pi[36] = {0,0,0,0,0,0,0,0, 1,1,1,1,1,1,1, 2,2,2,2,2,2,
                             3,3,3,3,3, 4,4,4,4, 5,5,5, 6,6, 7};
__constant__ int g_pj[36] = {0,1,2,3,4,5,6,7, 1,2,3,4,5,6,7, 2,3,4,5,6,7,
                             3,4,5,6,7, 4,5,6,7, 5,6,7, 6,7, 7};

__device__ __forceinline__ v8f wmma_f32x4(v2f a, v2f b, v8f c) {
  // D(16x16,f32) = A(16x4,f32) * B(4x16,f32) + C
  return __builtin_amdgcn_wmma_f32_16x16x4_f32(false, a, false, b,
                                               (short)0, c, false, false);
}

__device__ __forceinline__ float wave_sum(float v) {
#pragma unroll
  for (int off = 16; off > 0; off >>= 1) v += __shfl_xor(v, off, 32);
  return v;
}

__device__ __forceinline__ void sp_jet_base(float z, float& sp, float& s, float& sd) {
  s  = 1.f / (1.f + expf(-z));
  sp = fmaxf(z, 0.f) + log1pf(expf(-fabsf(z)));
  sd = s * (1.f - s);
}

// ---------------- GEMM: Z[c][o] = sum_k H[c][k] * W[o][k] -------------------
__device__ __forceinline__ void jet_gemm(const float* __restrict__ Wg,
                                         const float* __restrict__ Hs,
                                         float* __restrict__ Zs,
                                         int lane, int wave) {
  const int n0    = wave * 64;           // this wave's 4 N-tiles
  const int arow  = lane & 15;
  const int khalf = (lane >> 4) << 1;    // lanes 0-15: K 0,1   lanes 16-31: K 2,3
  const v8f vzero = {0.f,0.f,0.f,0.f,0.f,0.f,0.f,0.f};

  v8f acc[3][4];
#pragma unroll
  for (int mt = 0; mt < 3; mt++)
#pragma unroll
    for (int nt = 0; nt < 4; nt++) acc[mt][nt] = vzero;

  const float* habase = Hs + arow * LDSK + khalf;
  const float* wbase  = Wg + (n0 + arow) * HDIM + khalf;

#pragma unroll 2
  for (int k0 = 0; k0 < HDIM; k0 += 4) {
    v2f a[3];
#pragma unroll
    for (int mt = 0; mt < 3; mt++)
      a[mt] = *(const v2f*)(habase + mt * 16 * LDSK + k0);   // ds_load_b64
    v2f bf[4];
#pragma unroll
    for (int nt = 0; nt < 4; nt++)
      bf[nt] = *(const v2f*)(wbase + nt * 16 * HDIM + k0);   // global_load_b64
#pragma unroll
    for (int mt = 0; mt < 3; mt++)
#pragma unroll
      for (int nt = 0; nt < 4; nt++)
        acc[mt][nt] = wmma_f32x4(a[mt], bf[nt], acc[mt][nt]);
  }

  // C/D layout: vgpr r -> M=r (lanes 0-15) / M=r+8 (lanes 16-31), N = lane&15
  const int r8  = (lane >> 4) << 3;
  const int col = lane & 15;
#pragma unroll
  for (int mt = 0; mt < 3; mt++)
#pragma unroll
    for (int nt = 0; nt < 4; nt++)
#pragma unroll
      for (int r = 0; r < 8; r++)
        Zs[(mt * 16 + r8 + r) * LDSK + n0 + nt * 16 + col] = acc[mt][nt][r];
}

// ---------------- pointwise softplus jet (layers 1,2) -----------------------
__device__ __forceinline__ void softplus_stage(float* __restrict__ Zs,
                                               const float* __restrict__ bb,
                                               int tid) {
  for (int o = tid; o < HDIM; o += 256) {
    float z0 = Zs[o] + bb[o];
    float sp, s, sd;
    sp_jet_base(z0, sp, s, sd);
    float z1[8];
#pragma unroll
    for (int i = 0; i < 8; i++) z1[i] = Zs[(1 + i) * LDSK + o];
    Zs[o] = sp;
#pragma unroll
    for (int i = 0; i < 8; i++) Zs[(1 + i) * LDSK + o] = s * z1[i];
    int p = 0;
#pragma unroll
    for (int i = 0; i < 8; i++)
#pragma unroll
      for (int j = i; j < 8; j++) {
        float z2 = Zs[(9 + p) * LDSK + o];
        Zs[(9 + p) * LDSK + o] = s * z2 + sd * z1[i] * z1[j];
        p++;
      }
  }
}

// ---------------- LayerNorm jet: Z -> H -------------------------------------
__device__ __forceinline__ void ln_stage(float* __restrict__ Zs,
                                         float* __restrict__ Hs,
                                         const float* __restrict__ g,
                                         const float* __restrict__ be,
                                         float* __restrict__ mj,
                                         float* __restrict__ vj,
                                         float* __restrict__ rj,
                                         int tid) {
  const int lane = tid & 31, wave = tid >> 5;

  // channel means (45 reductions over 512, wave-per-channel)
  for (int c = wave; c < NCH; c += 8) {
    float s = 0.f;
    for (int k = lane; k < HDIM; k += 32) s += Zs[c * LDSK + k];
    s = wave_sum(s);
    if (lane == 0) mj[c] = s * (1.f / HDIM);
  }
  __syncthreads();

  // variance jet: v = mean(d*d) with d = h - m
  for (int c = wave; c < NCH; c += 8) {
    float s = 0.f;
    const float m0 = mj[0];
    if (c == 0) {
      for (int k = lane; k < HDIM; k += 32) { float d = Zs[k] - m0; s += d * d; }
    } else if (c < 9) {
      const float mc = mj[c];
      for (int k = lane; k < HDIM; k += 32) {
        float d0 = Zs[k] - m0;
        float d1 = Zs[c * LDSK + k] - mc;
        s += 2.f * d0 * d1;
      }
    } else {
      const int i = g_pi[c - 9], j = g_pj[c - 9];
      const float mc = mj[c], mi = mj[1 + i], mjj = mj[1 + j];
      for (int k = lane; k < HDIM; k += 32) {
        float d0 = Zs[k] - m0;
        float d2 = Zs[c * LDSK + k] - mc;
        float di = Zs[(1 + i) * LDSK + k] - mi;
        float dj = Zs[(1 + j) * LDSK + k] - mjj;
        s += 2.f * (d0 * d2 + di * dj);
      }
    }
    s = wave_sum(s);
    if (lane == 0) vj[c] = s * (1.f / HDIM);
  }
  __syncthreads();

  // r = (v+eps)^(-1/2) jet (scalar; one thread)
  if (tid == 0) {
    float u  = vj[0] + LNEPS;
    float r0 = 1.f / sqrtf(u);
    float iu = 1.f / u;
    float A1 = -0.5f * r0 * iu;        // phi'
    float A2 = 0.75f * r0 * iu * iu;   // phi''
    rj[0] = r0;
#pragma unroll
    for (int i = 0; i < 8; i++) rj[1 + i] = A1 * vj[1 + i];
    int p = 0;
#pragma unroll
    for (int i = 0; i < 8; i++)
#pragma unroll
      for (int j = i; j < 8; j++) {
        rj[9 + p] = A1 * vj[9 + p] + A2 * vj[1 + i] * vj[1 + j];
        p++;
      }
  }
  __syncthreads();

  // y = (d * r) * g + be   (jet product of vector d with scalar jet r)
  for (int o = tid; o < HDIM; o += 256) {
    const float gg = g[o], bb = be[o];
    const float r0 = rj[0];
    float d0 = Zs[o] - mj[0];
    float d1[8], r1[8];
#pragma unroll
    for (int i = 0; i < 8; i++) {
      d1[i] = Zs[(1 + i) * LDSK + o] - mj[1 + i];
      r1[i] = rj[1 + i];
    }
    Hs[o] = d0 * r0 * gg + bb;
#pragma unroll
    for (int i = 0; i < 8; i++)
      Hs[(1 + i) * LDSK + o] = (d0 * r1[i] + d1[i] * r0) * gg;
    int p = 0;
#pragma unroll
    for (int i = 0; i < 8; i++)
#pragma unroll
      for (int j = i; j < 8; j++) {
        float d2 = Zs[(9 + p) * LDSK + o] - mj[9 + p];
        Hs[(9 + p) * LDSK + o] =
            (d0 * rj[9 + p] + d2 * r0 + d1[i] * r1[j] + d1[j] * r1[i]) * gg;
        p++;
      }
  }
  __syncthreads();
}

// ---------------------------------------------------------------------------
__global__ __launch_bounds__(256)
void lnn_jet_wmma_kernel(const float* __restrict__ x,
                         const float* __restrict__ W0, const float* __restrict__ b0,
                         const float* __restrict__ g0, const float* __restrict__ be0,
                         const float* __restrict__ W1, const float* __restrict__ b1,
                         const float* __restrict__ g1, const float* __restrict__ be1,
                         const float* __restrict__ W2, const float* __restrict__ b2,
                         const float* __restrict__ g2, const float* __restrict__ be2,
                         const float* __restrict__ W3,
                         float* __restrict__ out) {
  extern __shared__ float smem[];
  float* Hs = smem;                       // [48][516] jet GEMM input
  float* Zs = smem + MPAD * LDSK;         // [48][516] jet GEMM output
  float* mj = smem + 2 * MPAD * LDSK;     // 48
  float* vj = mj + 48;                    // 48
  float* rj = vj + 48;                    // 48
  float* fj = rj + 48;                    // 48

  const int bi   = blockIdx.x;
  const int tid  = threadIdx.x;
  const int lane = tid & 31;
  const int wave = tid >> 5;

  // zero the 3 padding channel rows of H (GEMM A-matrix rows 45..47)
  for (int idx = tid; idx < 3 * LDSK; idx += 256) Hs[NCH * LDSK + idx] = 0.f;

  // ---- layer 0: z = W0 x + b0 (z1 = W0 row, z2 = 0), fused softplus jet ----
  float xr[8];
#pragma unroll
  for (int i = 0; i < 8; i++) xr[i] = x[bi * 8 + i];

  for (int o = tid; o < HDIM; o += 256) {
    float w[8];
    float z0 = b0[o];
#pragma unroll
    for (int i = 0; i < 8; i++) { w[i] = W0[o * 8 + i]; z0 = fmaf(w[i], xr[i], z0); }
    float sp, s, sd;
    sp_jet_base(z0, sp, s, sd);
    Zs[o] = sp;
#pragma unroll
    for (int i = 0; i < 8; i++) Zs[(1 + i) * LDSK + o] = s * w[i];
    int p = 0;
#pragma unroll
    for (int i = 0; i < 8; i++)
#pragma unroll
      for (int j = i; j < 8; j++) {
        Zs[(9 + p) * LDSK + o] = sd * w[i] * w[j];
        p++;
      }
  }
  __syncthreads();
  ln_stage(Zs, Hs, g0, be0, mj, vj, rj, tid);

  // ---- layer 1 ----
  jet_gemm(W1, Hs, Zs, lane, wave);
  __syncthreads();
  softplus_stage(Zs, b1, tid);
  __syncthreads();
  ln_stage(Zs, Hs, g1, be1, mj, vj, rj, tid);

  // ---- layer 2 ----
  jet_gemm(W2, Hs, Zs, lane, wave);
  __syncthreads();
  softplus_stage(Zs, b2, tid);
  __syncthreads();
  ln_stage(Zs, Hs, g2, be2, mj, vj, rj, tid);

  // ---- final layer: f_c = sum_k W3[k] * H[c][k]  (bias irrelevant for derivs)
  for (int c = wave; c < NCH; c += 8) {
    float s = 0.f;
    for (int k = lane; k < HDIM; k += 32) s += Hs[c * LDSK + k] * W3[k];
    s = wave_sum(s);
    if (lane == 0) fj[c] = s;
  }
  __syncthreads();

  // ---- assemble Hessian, solve 4x4 system (thread 0, scratch in LDS) ------
  if (tid == 0) {
    float* Hm = Zs;        // 8x8
    float* MM = Zs + 64;   // 4x5 augmented
    int p = 0;
#pragma unroll
    for (int i = 0; i < 8; i++)
#pragma unroll
      for (int j = i; j < 8; j++) {
        float h = fj[9 + p];
        Hm[i * 8 + j] = h;
        Hm[j * 8 + i] = h;
        p++;
      }
    float xd[4];
#pragma unroll
    for (int j = 0; j < 4; j++) xd[j] = x[bi * 8 + 4 + j];
    // rhs_k = J_k - sum_j Hm[j][4+k] * qd_j ;  A[k][j] = Hm[4+k][4+j]
    for (int k = 0; k < 4; k++) {
      float t = 0.f;
      for (int j = 0; j < 4; j++) t = fmaf(Hm[j * 8 + 4 + k], xd[j], t);
      MM[k * 5 + 4] = fj[1 + k] - t;
      for (int j = 0; j < 4; j++) MM[k * 5 + j] = Hm[(4 + k) * 8 + 4 + j];
    }
    // Gauss-Jordan with partial pivoting
    for (int col = 0; col < 4; col++) {
      int piv = col;
      float best = fabsf(MM[col * 5 + col]);
      for (int r = col + 1; r < 4; r++) {
        float a = fabsf(MM[r * 5 + col]);
        if (a > best) { best = a; piv = r; }
      }
      if (piv != col)
        for (int cc = 0; cc < 5; cc++) {
          float t = MM[col * 5 + cc];
          MM[col * 5 + cc] = MM[piv * 5 + cc];
          MM[piv * 5 + cc] = t;
        }
      float inv = 1.f / MM[col * 5 + col];
      for (int cc = 0; cc < 5; cc++) MM[col * 5 + cc] *= inv;
      for (int r = 0; r < 4; r++)
        if (r != col) {
          float f = MM[r * 5 + col];
          for (int cc = 0; cc < 5; cc++) MM[r * 5 + cc] -= f * MM[col * 5 + cc];
        }
    }
#pragma unroll
    for (int k = 0; k < 4; k++) out[bi * 4 + k] = MM[k * 5 + 4];
  }
}

// ---------------------------------------------------------------------------
extern "C" void kernel_launch(void* const* d_in, const int* in_sizes, int n_in,
                              void* d_out, int out_size, void* d_ws, size_t ws_size,
                              hipStream_t stream) {
  (void)n_in; (void)out_size; (void)d_ws; (void)ws_size;
  const float* x   = (const float*)d_in[0];
  const float* W0  = (const float*)d_in[1];
  const float* b0  = (const float*)d_in[2];
  const float* g0  = (const float*)d_in[3];
  const float* be0 = (const float*)d_in[4];
  const float* W1  = (const float*)d_in[5];
  const float* b1  = (const float*)d_in[6];
  const float* g1  = (const float*)d_in[7];
  const float* be1 = (const float*)d_in[8];
  const float* W2  = (const float*)d_in[9];
  const float* b2  = (const float*)d_in[10];
  const float* g2  = (const float*)d_in[11];
  const float* be2 = (const float*)d_in[12];
  const float* W3  = (const float*)d_in[13];
  // d_in[14] = b3: constant offset, no effect on J/H -> unused

  const int B = in_sizes[0] / 8;
  const size_t smem = (size_t)(2 * MPAD * LDSK + 4 * 48) * sizeof(float);
  hipFuncSetAttribute((const void*)lnn_jet_wmma_kernel,
                      hipFuncAttributeMaxDynamicSharedMemorySize, (int)smem);
  lnn_jet_wmma_kernel<<<B, 256, smem, stream>>>(
      x, W0, b0, g0, be0, W1, b1, g1, be1, W2, b2, g2, be2, W3, (float*)d_out);
}